// GATClassifier_53996328846047
// MI455X (gfx1250) — compile-verified
//
#include <hip/hip_runtime.h>

#define NN   40000
#define EE   640000
#define EPAD 680000      // E + N self-loops
#define FIN  32
#define HH   8
#define CCH  16
#define HC   128
#define GG   1000
#define MP   1008        // GG padded up to a multiple of 16 (63 row tiles)
#define MMLP 256
#define NEG_SLOPE 0.2f

typedef __attribute__((ext_vector_type(2))) float v2f;
typedef __attribute__((ext_vector_type(8))) float v8f;

// ---------------------------------------------------------------------------
// FP32 WMMA GEMM, TN form: D[M,N] = A[M,K] @ Bt[N,K]^T (+bias, optional ReLU).
// One wave per 16x16 tile; K statically unrolled. With B pre-transposed, both
// the A and B operands for v_wmma_f32_16x16x4_f32 are contiguous float pairs
// (lanes 0-15 take K+0/K+1, lanes 16-31 take K+2/K+3 per ISA 7.12.2), so every
// operand fetch is a coalesced global_load_b64 and there are no EXEC guards.
// M, N must be multiples of 16 (callers pad).
// ---------------------------------------------------------------------------
template <int KDIM>
__global__ void wmma_gemm_tn(const float* __restrict__ A, const float* __restrict__ Bt,
                             const float* __restrict__ bias, float* __restrict__ D,
                             int Ndim, int relu) {
  const int lane = threadIdx.x & 31;
  const int m    = lane & 15;
  const int kh   = (lane >> 4) << 1;          // 0 or 2
  const float* Ap = A  + (size_t)(blockIdx.x * 16 + m) * KDIM + kh;
  const float* Bp = Bt + (size_t)(blockIdx.y * 16 + m) * KDIM + kh;

  v8f acc = {};
#pragma unroll
  for (int k = 0; k < KDIM; k += 4) {
    v2f a = *(const v2f*)(Ap + k);
    v2f b = *(const v2f*)(Bp + k);
    acc = __builtin_amdgcn_wmma_f32_16x16x4_f32(false, a, false, b,
                                                (short)0, acc, false, false);
  }

  const int bcol  = blockIdx.y * 16 + m;
  const int rbase = blockIdx.x * 16 + ((lane >> 4) << 3);
  const float bb = bias ? bias[bcol] : 0.f;
#pragma unroll
  for (int r = 0; r < 8; ++r) {
    float v = acc[r] + bb;
    if (relu) v = v > 0.f ? v : 0.f;
    D[(size_t)(rbase + r) * Ndim + bcol] = v;
  }
}

// out[n,k] = in[k,n] : one-shot weight transpose so GEMM B-operand is contiguous
__global__ void transpose_f32(const float* __restrict__ in, float* __restrict__ out,
                              int K, int N) {
  int i = blockIdx.x * blockDim.x + threadIdx.x;
  if (i >= K * N) return;
  int k = i / N, n = i - k * N;
  out[(size_t)n * K + k] = in[i];
}

// Per-(node,head) attention logit dot products: al = <h[n,h,:], a[h,:]>
__global__ void attn_logits(const float* __restrict__ h, const float* __restrict__ a_src,
                            const float* __restrict__ a_dst,
                            float* __restrict__ alS, float* __restrict__ alD) {
  int idx = blockIdx.x * blockDim.x + threadIdx.x;
  if (idx >= NN * HH) return;
  int head = idx & (HH - 1);
  const float* hp = h + (size_t)(idx >> 3) * HC + head * CCH;
  const float* as = a_src + head * CCH;
  const float* ad = a_dst + head * CCH;
  float s = 0.f, d = 0.f;
#pragma unroll
  for (int c = 0; c < CCH; ++c) { float v = hp[c]; s += v * as[c]; d += v * ad[c]; }
  alS[idx] = s;
  alD[idx] = d;
}

__global__ void fill_f32(float* __restrict__ p, float v, int n) {
  int i = blockIdx.x * blockDim.x + threadIdx.x;
  if (i < n) p[i] = v;
}

// float atomic max via signed/unsigned integer ordering trick (init = -inf)
__device__ __forceinline__ void atomicMaxF(float* addr, float val) {
  if (val >= 0.f) atomicMax((int*)addr, __float_as_int(val));
  else            atomicMin((unsigned int*)addr, __float_as_uint(val));
}

// Edge pass 1: e = leakyrelu(alS[src] + alD[dst]); segment-max over dst
__global__ void edge_logit_max(const int* __restrict__ esrc, const int* __restrict__ edst,
                               const float* __restrict__ alS, const float* __restrict__ alD,
                               float* __restrict__ ebuf, float* __restrict__ emax) {
  int idx = blockIdx.x * blockDim.x + threadIdx.x;
  if (idx >= EPAD * HH) return;
  int e = idx >> 3, head = idx & (HH - 1);
  int s = (e < EE) ? esrc[e] : (e - EE);
  int d = (e < EE) ? edst[e] : (e - EE);
  float v = alS[s * HH + head] + alD[d * HH + head];
  v = (v > 0.f) ? v : NEG_SLOPE * v;
  ebuf[idx] = v;
  atomicMaxF(&emax[d * HH + head], v);
}

// Edge pass 2: ex = exp(e - emax[dst]); segment-sum into denom
__global__ void edge_exp_sum(const int* __restrict__ edst,
                             float* __restrict__ ebuf, const float* __restrict__ emax,
                             float* __restrict__ denom) {
  int idx = blockIdx.x * blockDim.x + threadIdx.x;
  if (idx >= EPAD * HH) return;
  int e = idx >> 3, head = idx & (HH - 1);
  int d = (e < EE) ? edst[e] : (e - EE);
  float ex = __expf(ebuf[idx] - emax[d * HH + head]);
  ebuf[idx] = ex;
  atomicAdd(&denom[d * HH + head], ex);
}

// Edge pass 3: out[dst] += (ex/denom[dst]) * h[src]; one thread per (edge, channel).
// h and out (~20 MB each) stay resident in MI455X's 192 MB L2, so this gather
// + f32 atomic scatter runs at L2 rate, not HBM.
__global__ void edge_aggregate(const int* __restrict__ esrc, const int* __restrict__ edst,
                               const float* __restrict__ ebuf, const float* __restrict__ denom,
                               const float* __restrict__ h, float* __restrict__ out) {
  unsigned idx = blockIdx.x * blockDim.x + threadIdx.x;
  if (idx >= (unsigned)EPAD * HC) return;
  unsigned e = idx >> 7;
  int c = idx & (HC - 1);
  int head = c >> 4;
  int s = (e < EE) ? esrc[e] : (int)(e - EE);
  int d = (e < EE) ? edst[e] : (int)(e - EE);
  float alpha = ebuf[(size_t)e * HH + head] / denom[d * HH + head];
  atomicAdd(&out[(size_t)d * HC + c], alpha * h[(size_t)s * HC + c]);
}

__global__ void bias_relu_inplace(float* __restrict__ x, const float* __restrict__ b) {
  int i = blockIdx.x * blockDim.x + threadIdx.x;
  if (i >= NN * HC) return;
  float v = x[i] + b[i & (HC - 1)];
  x[i] = v > 0.f ? v : 0.f;
}

__global__ void pool_acc(const float* __restrict__ h, const int* __restrict__ batch,
                         float* __restrict__ pool, float* __restrict__ cnt) {
  int i = blockIdx.x * blockDim.x + threadIdx.x;
  if (i >= NN * HC) return;
  int n = i >> 7, c = i & (HC - 1);
  int g = batch[n];
  atomicAdd(&pool[(size_t)g * HC + c], h[i]);
  if (c == 0) atomicAdd(&cnt[g], 1.0f);
}

__global__ void pool_div(float* __restrict__ pool, const float* __restrict__ cnt) {
  int i = blockIdx.x * blockDim.x + threadIdx.x;
  if (i >= GG * HC) return;
  float c = cnt[i >> 7];
  pool[i] /= (c > 1.f ? c : 1.f);
}

// Final 256 -> 2 projection: one thread per (graph, output)
__global__ void head_out(const float* __restrict__ g, const float* __restrict__ oW,
                         const float* __restrict__ ob, float* __restrict__ out) {
  int i = blockIdx.x * blockDim.x + threadIdx.x;
  if (i >= GG * 2) return;
  int gi = i >> 1, o = i & 1;
  float acc = ob[o];
  const float* gp = g + (size_t)gi * MMLP;
#pragma unroll 8
  for (int k = 0; k < MMLP; ++k) acc += gp[k] * oW[k * 2 + o];
  out[i] = acc;
}

static inline int cdiv_ll(long long a, long long b) { return (int)((a + b - 1) / b); }

extern "C" void kernel_launch(void* const* d_in, const int* in_sizes, int n_in,
                              void* d_out, int out_size, void* d_ws, size_t ws_size,
                              hipStream_t stream) {
  (void)in_sizes; (void)n_in; (void)out_size; (void)ws_size;

  const float* x    = (const float*)d_in[0];
  const int*   eidx = (const int*)d_in[1];
  const int*   esrc = eidx;            // edge_index[0]
  const int*   edst = eidx + EE;       // edge_index[1]
  const int*   batch= (const int*)d_in[2];
  const float* W[3]  = {(const float*)d_in[3],  (const float*)d_in[7],  (const float*)d_in[11]};
  const float* b[3]  = {(const float*)d_in[4],  (const float*)d_in[8],  (const float*)d_in[12]};
  const float* as[3] = {(const float*)d_in[5],  (const float*)d_in[9],  (const float*)d_in[13]};
  const float* ad[3] = {(const float*)d_in[6],  (const float*)d_in[10], (const float*)d_in[14]};
  const float* fW0 = (const float*)d_in[15];
  const float* fb0 = (const float*)d_in[16];
  const float* fW1 = (const float*)d_in[17];
  const float* fb1 = (const float*)d_in[18];
  const float* oW  = (const float*)d_in[19];
  const float* ob  = (const float*)d_in[20];
  float* out = (float*)d_out;

  // workspace carve-up (floats); ~72 MB total
  float* p = (float*)d_ws;
  float* hA    = p; p += (size_t)NN * HC;       // transformed features
  float* hB    = p; p += (size_t)NN * HC;       // aggregation accumulator / layer output
  float* alS   = p; p += NN * HH;
  float* alD   = p; p += NN * HH;
  float* emax  = p; p += NN * HH;
  float* denom = p; p += NN * HH;
  float* ebuf  = p; p += (size_t)EPAD * HH;
  float* pool  = p; p += (size_t)MP * HC;       // padded to 1008 rows
  float* cnt   = p; p += 1024;
  float* m0    = p; p += (size_t)MP * MMLP;     // padded
  float* m1    = p; p += (size_t)MP * MMLP;     // padded
  float* Wt[3];
  Wt[0] = p; p += FIN * HC;                     // W0^T  [128 x 32]
  Wt[1] = p; p += HC * HC;                      // W1^T  [128 x 128]
  Wt[2] = p; p += HC * HC;                      // W2^T  [128 x 128]
  float* fWt0 = p; p += HC * MMLP;              // fW0^T [256 x 128]
  float* fWt1 = p; p += MMLP * MMLP;            // fW1^T [256 x 256]

  const int TB = 256;
  const float NEG_INF = -__builtin_huge_valf();

  // one-shot weight transposes (N-major) so WMMA B-operands are b64 loads
  transpose_f32<<<cdiv_ll(FIN * HC, TB),   TB, 0, stream>>>(W[0], Wt[0], FIN, HC);
  transpose_f32<<<cdiv_ll(HC * HC, TB),    TB, 0, stream>>>(W[1], Wt[1], HC, HC);
  transpose_f32<<<cdiv_ll(HC * HC, TB),    TB, 0, stream>>>(W[2], Wt[2], HC, HC);
  transpose_f32<<<cdiv_ll(HC * MMLP, TB),  TB, 0, stream>>>(fW0, fWt0, HC, MMLP);
  transpose_f32<<<cdiv_ll(MMLP * MMLP, TB),TB, 0, stream>>>(fW1, fWt1, MMLP, MMLP);

  const float* in = x;
  for (int L = 0; L < 3; ++L) {
    dim3 gt(NN / 16, HC / 16);
    if (L == 0)
      wmma_gemm_tn<FIN><<<gt, 32, 0, stream>>>(in, Wt[0], nullptr, hA, HC, 0);
    else
      wmma_gemm_tn<HC><<<gt, 32, 0, stream>>>(in, Wt[L], nullptr, hA, HC, 0);
    attn_logits<<<cdiv_ll(NN * HH, TB), TB, 0, stream>>>(hA, as[L], ad[L], alS, alD);
    fill_f32<<<cdiv_ll(NN * HH, TB), TB, 0, stream>>>(emax, NEG_INF, NN * HH);
    fill_f32<<<cdiv_ll(NN * HH, TB), TB, 0, stream>>>(denom, 0.f, NN * HH);
    fill_f32<<<cdiv_ll((long long)NN * HC, TB), TB, 0, stream>>>(hB, 0.f, NN * HC);
    edge_logit_max<<<cdiv_ll((long long)EPAD * HH, TB), TB, 0, stream>>>(esrc, edst, alS, alD, ebuf, emax);
    edge_exp_sum<<<cdiv_ll((long long)EPAD * HH, TB), TB, 0, stream>>>(edst, ebuf, emax, denom);
    edge_aggregate<<<cdiv_ll((long long)EPAD * HC, TB), TB, 0, stream>>>(esrc, edst, ebuf, denom, hA, hB);
    bias_relu_inplace<<<cdiv_ll((long long)NN * HC, TB), TB, 0, stream>>>(hB, b[L]);
    in = hB;
  }

  // global mean pool (pool buffer zeroed through the padded rows)
  fill_f32<<<cdiv_ll((long long)MP * HC, TB), TB, 0, stream>>>(pool, 0.f, MP * HC);
  fill_f32<<<cdiv_ll(GG, TB), TB, 0, stream>>>(cnt, 0.f, GG);
  pool_acc<<<cdiv_ll((long long)NN * HC, TB), TB, 0, stream>>>(hB, batch, pool, cnt);
  pool_div<<<cdiv_ll(GG * HC, TB), TB, 0, stream>>>(pool, cnt);

  // MLP head: padded-M unguarded WMMA GEMMs with fused bias+ReLU
  dim3 g0(MP / 16, MMLP / 16);
  wmma_gemm_tn<HC><<<g0, 32, 0, stream>>>(pool, fWt0, fb0, m0, MMLP, 1);
  wmma_gemm_tn<MMLP><<<g0, 32, 0, stream>>>(m0, fWt1, fb1, m1, MMLP, 1);
  head_out<<<cdiv_ll(GG * 2, TB), TB, 0, stream>>>(m1, oW, ob, out);
}